// Model_83519934038714
// MI455X (gfx1250) — compile-verified
//
#include <hip/hip_runtime.h>
#include <hip/hip_bf16.h>

typedef float v2f __attribute__((ext_vector_type(2)));
typedef float v8f __attribute__((ext_vector_type(8)));

#define IN_F 128
#define HIDF 64
#define OUTF 40
#define TOLF 1e-5f

__device__ __forceinline__ void atomic_add_f32(float* p, float v) {
  __hip_atomic_fetch_add(p, v, __ATOMIC_RELAXED, __HIP_MEMORY_SCOPE_AGENT);
}

// ---------------- zero init (ctrl + u + acc are contiguous) ----------------
__global__ void k_zero(float* __restrict__ p, long n) {
  long i = (long)blockIdx.x * blockDim.x + threadIdx.x;
  long stride = (long)gridDim.x * blockDim.x;
  for (; i < n; i += stride) p[i] = 0.0f;
}

// ---------------- Wc = W_bias @ W_enc  [64,128]  (tiny) ----------------
__global__ void k_wc(const float* __restrict__ Wb, const float* __restrict__ We,
                     float* __restrict__ Wc) {
  int idx = blockIdx.x * blockDim.x + threadIdx.x;
  if (idx >= HIDF * IN_F) return;
  int r = idx >> 7, c = idx & (IN_F - 1);
  float s = 0.f;
#pragma unroll 8
  for (int k = 0; k < HIDF; ++k) s = fmaf(Wb[r * HIDF + k], We[k * IN_F + c], s);
  Wc[idx] = s;
}

// ---------------- bias = X @ Wc^T  via V_WMMA_F32_16X16X4_F32 ----------------
// one wave per 16x16 output tile; blockDim=128 -> 4 waves = 4 column tiles
__global__ void __launch_bounds__(128)
k_gemm_bias(const float* __restrict__ X, const float* __restrict__ Wc,
            float* __restrict__ Y, int nrows) {
  int lane = threadIdx.x & 31;
  int wave = threadIdx.x >> 5;        // column tile 0..3
  int half = lane >> 4;               // 0: K={0,1}, 1: K={2,3}
  int m    = lane & 15;
  int row0 = blockIdx.x * 16;

  int arow = row0 + m; if (arow >= nrows) arow = nrows - 1;
  const float* xr = X  + (size_t)arow * IN_F;
  const float* wr = Wc + (size_t)(wave * 16 + m) * IN_F;

  v8f c = {};
#pragma unroll 4
  for (int k = 0; k < IN_F; k += 4) {
    v2f a = *(const v2f*)(xr + k + 2 * half);
    v2f b = *(const v2f*)(wr + k + 2 * half);
    c = __builtin_amdgcn_wmma_f32_16x16x4_f32(false, a, false, b, (short)0, c,
                                              false, false);
  }
  float* yb = Y + (size_t)row0 * HIDF + wave * 16;
#pragma unroll
  for (int r = 0; r < 8; ++r) {
    int row = half * 8 + r;
    if (row0 + row < nrows) yb[(size_t)row * HIDF + m] = c[r];
  }
}

// ---------------- SpMM: acc[dst] += w * u[src]  (16 lanes per edge) --------
__global__ void __launch_bounds__(256)
k_prop(const int* __restrict__ ei, const float* __restrict__ ew,
       const float* __restrict__ u, float* __restrict__ acc,
       const int* __restrict__ done, int gated, int E) {
  if (gated && *done) return;
  int t = blockIdx.x * blockDim.x + threadIdx.x;
  if (t >= E * 16) return;
  int e = t >> 4;
  int q = (t & 15) << 2;            // 4 consecutive features
  int s = ei[e];                    // edge_index[0][e]
  int d = ei[E + e];                // edge_index[1][e]
  float w = ew[e];
  float4 uv = *(const float4*)(u + (size_t)s * HIDF + q);
  float* ap = acc + (size_t)d * HIDF + q;
  atomic_add_f32(ap + 0, w * uv.x);
  atomic_add_f32(ap + 1, w * uv.y);
  atomic_add_f32(ap + 2, w * uv.z);
  atomic_add_f32(ap + 3, w * uv.w);
}

// ---- Picard step (in place), acc re-zeroed, partial norms accumulated -----
__global__ void __launch_bounds__(256)
k_step(float* __restrict__ u, float* __restrict__ acc,
       const float* __restrict__ bias, float* __restrict__ sums,
       const int* __restrict__ done, int gated,
       const float* __restrict__ beta_p, const float* __restrict__ gamma_p,
       int n4) {
  if (gated && *done) return;
  const float beta = *beta_p, gamma = *gamma_p;
  int t = blockIdx.x * blockDim.x + threadIdx.x;
  float d2 = 0.f, n2 = 0.f;
  if (t < n4) {
    float4 a  = ((float4*)acc)[t];
    float4 b  = ((const float4*)bias)[t];
    float4 uo = ((float4*)u)[t];
    float4 un;
    un.x = (1.f - beta) * uo.x + beta * fmaxf(fmaf(gamma, a.x, b.x), 0.f);
    un.y = (1.f - beta) * uo.y + beta * fmaxf(fmaf(gamma, a.y, b.y), 0.f);
    un.z = (1.f - beta) * uo.z + beta * fmaxf(fmaf(gamma, a.z, b.z), 0.f);
    un.w = (1.f - beta) * uo.w + beta * fmaxf(fmaf(gamma, a.w, b.w), 0.f);
    float dx = un.x - uo.x, dy = un.y - uo.y, dz = un.z - uo.z, dw = un.w - uo.w;
    d2 = dx * dx + dy * dy + dz * dz + dw * dw;
    n2 = uo.x * uo.x + uo.y * uo.y + uo.z * uo.z + uo.w * uo.w;
    ((float4*)u)[t] = un;
    ((float4*)acc)[t] = make_float4(0.f, 0.f, 0.f, 0.f);
  }
  __shared__ float sd[256], sn[256];
  int lt = threadIdx.x;
  sd[lt] = d2; sn[lt] = n2;
  __syncthreads();
  for (int off = 128; off > 0; off >>= 1) {
    if (lt < off) { sd[lt] += sd[lt + off]; sn[lt] += sn[lt + off]; }
    __syncthreads();
  }
  if (lt == 0) {
    atomic_add_f32(&sums[0], sd[0]);
    atomic_add_f32(&sums[1], sn[0]);
  }
}

// ---------------- convergence check ----------------
__global__ void k_check(float* __restrict__ sums, int* __restrict__ done) {
  if (*done) return;
  float err = sqrtf(sums[0]) / (sqrtf(sums[1]) + 1e-9f);
  if (err <= TOLF) *done = 1;
  sums[0] = 0.f; sums[1] = 0.f;
}

// ---------------- out = relu(u) @ W_dec^T  (WMMA, 3 col tiles, mask 40) ----
__global__ void __launch_bounds__(96)
k_decode(const float* __restrict__ U, const float* __restrict__ Wd,
         float* __restrict__ Out, int nrows) {
  int lane = threadIdx.x & 31;
  int wave = threadIdx.x >> 5;        // column tile 0..2 (cols padded to 48)
  int half = lane >> 4;
  int m    = lane & 15;
  int row0 = blockIdx.x * 16;
  int n    = wave * 16 + m;

  int arow = row0 + m; if (arow >= nrows) arow = nrows - 1;
  const float* ur = U + (size_t)arow * HIDF;
  const float* wr = Wd + (size_t)(n < OUTF ? n : 0) * HIDF;
  bool nvalid = (n < OUTF);

  v8f c = {};
#pragma unroll 4
  for (int k = 0; k < HIDF; k += 4) {
    v2f a, b;
    a.x = fmaxf(ur[k + 2 * half], 0.f);
    a.y = fmaxf(ur[k + 2 * half + 1], 0.f);
    v2f bl = *(const v2f*)(wr + k + 2 * half);
    b.x = nvalid ? bl.x : 0.f;
    b.y = nvalid ? bl.y : 0.f;
    c = __builtin_amdgcn_wmma_f32_16x16x4_f32(false, a, false, b, (short)0, c,
                                              false, false);
  }
  if (nvalid) {
#pragma unroll
    for (int r = 0; r < 8; ++r) {
      int row = row0 + half * 8 + r;
      if (row < nrows) Out[(size_t)row * OUTF + n] = c[r];
    }
  }
}

extern "C" void kernel_launch(void* const* d_in, const int* in_sizes, int n_in,
                              void* d_out, int out_size, void* d_ws, size_t ws_size,
                              hipStream_t stream) {
  const float* x       = (const float*)d_in[0];
  const int*   ei      = (const int*)d_in[1];
  const float* ew      = (const float*)d_in[2];
  // d_in[3] = batch (unused; all-zero graph id)
  const float* Wen     = (const float*)d_in[4];
  const float* Wb      = (const float*)d_in[5];
  const float* Wd      = (const float*)d_in[6];
  const float* beta_p  = (const float*)d_in[7];
  const float* gamma_p = (const float*)d_in[8];

  const int N = in_sizes[0] / IN_F;
  const int E = in_sizes[2];

  // workspace layout: [ctrl 256B][u N*64][acc N*64][bias N*64][Wc 64*128]
  char*  ws   = (char*)d_ws;
  float* sums = (float*)ws;                 // sums[0]=d2, sums[1]=u2
  int*   done = (int*)(ws + 128);
  size_t off  = 256;
  float* u    = (float*)(ws + off); off += (size_t)N * HIDF * sizeof(float);
  float* acc  = (float*)(ws + off); off += (size_t)N * HIDF * sizeof(float);
  float* bias = (float*)(ws + off); off += (size_t)N * HIDF * sizeof(float);
  float* Wc   = (float*)(ws + off);

  // zero ctrl + u + acc (contiguous region)
  long nz = 64 + 2L * N * HIDF;
  k_zero<<<2048, 256, 0, stream>>>((float*)ws, nz);

  // fused encoder+bias weights, then bias_x = x @ Wc^T (WMMA)
  k_wc<<<(HIDF * IN_F + 255) / 256, 256, 0, stream>>>(Wb, Wen, Wc);
  k_gemm_bias<<<(N + 15) / 16, 128, 0, stream>>>(x, Wc, bias, N);

  const int propBlocks = (int)(((long)E * 16 + 255) / 256);
  const int n4 = N * (HIDF / 4);
  const int stepBlocks = (n4 + 255) / 256;

  // gradient-free fixed-point solve: device-gated, max 100 iterations
  for (int it = 0; it < 100; ++it) {
    k_prop<<<propBlocks, 256, 0, stream>>>(ei, ew, u, acc, done, 1, E);
    k_step<<<stepBlocks, 256, 0, stream>>>(u, acc, bias, sums, done, 1,
                                           beta_p, gamma_p, n4);
    k_check<<<1, 1, 0, stream>>>(sums, done);
  }
  // 5 phantom Picard steps (forward math identical; always run)
  for (int p = 0; p < 5; ++p) {
    k_prop<<<propBlocks, 256, 0, stream>>>(ei, ew, u, acc, done, 0, E);
    k_step<<<stepBlocks, 256, 0, stream>>>(u, acc, bias, sums, done, 0,
                                           beta_p, gamma_p, n4);
  }

  // decode (WMMA) into d_out[0 : N*40], then append u as second output
  k_decode<<<(N + 15) / 16, 96, 0, stream>>>(u, Wd, (float*)d_out, N);
  hipMemcpyAsync((float*)d_out + (size_t)N * OUTF, u,
                 (size_t)N * HIDF * sizeof(float),
                 hipMemcpyDeviceToDevice, stream);
}